// ImageWarp_7971459301550
// MI455X (gfx1250) — compile-verified
//
#include <hip/hip_runtime.h>

// ---------------------------------------------------------------------------
// Problem constants (from reference): B=16, C=3, H=384, W=1280, fp32 in/out.
// ---------------------------------------------------------------------------
#define EPS_F 1e-7f
constexpr int Bn = 16, Cn = 3, Hn = 384, Wn = 1280;
constexpr int HWn = Hn * Wn;                       // 491520
constexpr int PIX_PER_THREAD = 4;                  // W % 4 == 0, groups never cross rows
constexpr int THREADS = 256;
constexpr int TOTAL_GROUPS = Bn * HWn / PIX_PER_THREAD;   // 1,966,080
constexpr int BLOCKS = TOTAL_GROUPS / THREADS;            // 7680 (exact)

typedef __attribute__((ext_vector_type(2))) float v2f;
typedef __attribute__((ext_vector_type(4))) float v4f;
typedef __attribute__((ext_vector_type(8))) float v8f;

// ---------------------------------------------------------------------------
// Setup kernel (1 wave): G(3x4) = K4 * T4 * J, J = [[Kinv33,0],[0,0,0,1]].
// Then per-pixel: cp = G * [x*d, y*d, d, 1].
// The two 4x4 products are done with V_WMMA_F32_16X16X4_F32 (D = A(16x4)*B(4x16)),
// embedding the 4x4 matrices in the top-left of the WMMA operands.
// f32 A-operand layout (ISA 7.12.2): lane L holds row M=L&15; VGPR0/1 = K = 2*(L>>4), +1.
// f32 B-operand layout (by the B-striping rule): lane L holds col N=L&15;
//                                               VGPR0/1 = K = 2*(L>>4), +1.
// f32 D layout: element m of v8f at lane n<16 = D[m][n].
// WMMA1 result is LDS-staged to re-shape D-layout -> B-operand layout for WMMA2.
// ---------------------------------------------------------------------------
__global__ __launch_bounds__(32) void warp_setup(
    const float* __restrict__ pose,   // [3,4]
    const float* __restrict__ Kb,     // [B,4,4] (batch-identical; use batch 0)
    const float* __restrict__ Kinv,   // [B,4,4] (batch-identical; use batch 0)
    float* __restrict__ G)            // out: 12 floats, G[m*4+n], m=0..2
{
#if __has_builtin(__builtin_amdgcn_wmma_f32_16x16x4_f32)
    __shared__ float lds[16];
    const int lane = threadIdx.x;          // 0..31
    const int row  = lane & 15;            // M (A) / N (B)
    const int kb   = (lane >> 4) * 2;      // first K slot this lane holds

    // T4 = [[pose],[0,0,0,1]] as A-operand; J as B-operand.
    v2f a1 = {0.0f, 0.0f}, b1 = {0.0f, 0.0f};
    if (row < 4) {
        // A: T4[row][kb], T4[row][kb+1]
        float t0 = (row < 3) ? pose[row * 4 + kb]     : (kb     == 3 ? 1.0f : 0.0f);
        float t1 = (row < 3) ? pose[row * 4 + kb + 1] : (kb + 1 == 3 ? 1.0f : 0.0f);
        a1.x = t0; a1.y = t1;
        // B: J[kb][row], J[kb+1][row]   (J top-left 3x3 = Kinv33, J[3][3]=1)
        float j0 = (kb     < 3 && row < 3) ? Kinv[kb * 4 + row]       : ((kb     == 3 && row == 3) ? 1.0f : 0.0f);
        float j1 = (kb + 1 < 3 && row < 3) ? Kinv[(kb + 1) * 4 + row] : ((kb + 1 == 3 && row == 3) ? 1.0f : 0.0f);
        b1.x = j0; b1.y = j1;
    }
    v8f cz = {};
    v8f d1 = __builtin_amdgcn_wmma_f32_16x16x4_f32(false, a1, false, b1,
                                                   (short)0, cz, false, false);
    // Stage W1 = T4*J (4x4) through LDS (D layout -> B-operand layout).
    if (lane < 4) {
        lds[0 * 4 + lane] = d1[0];
        lds[1 * 4 + lane] = d1[1];
        lds[2 * 4 + lane] = d1[2];
        lds[3 * 4 + lane] = d1[3];
    }
    __syncthreads();

    // WMMA2: G4 = K4 * W1
    v2f a2 = {0.0f, 0.0f}, b2 = {0.0f, 0.0f};
    if (row < 4) {
        a2.x = Kb[row * 4 + kb];
        a2.y = Kb[row * 4 + kb + 1];
        b2.x = lds[kb * 4 + row];
        b2.y = lds[(kb + 1) * 4 + row];
    }
    v8f d2 = __builtin_amdgcn_wmma_f32_16x16x4_f32(false, a2, false, b2,
                                                   (short)0, cz, false, false);
    if (lane < 4) {
        G[0 * 4 + lane] = d2[0];
        G[1 * 4 + lane] = d2[1];
        G[2 * 4 + lane] = d2[2];
    }
#else
    // Scalar fallback (kept for toolchain portability).
    if (threadIdx.x == 0) {
        float T4[4][4], J[4][4], W1[4][4];
        for (int r = 0; r < 4; ++r)
            for (int c = 0; c < 4; ++c) {
                T4[r][c] = (r < 3) ? pose[r * 4 + c] : (c == 3 ? 1.0f : 0.0f);
                J[r][c]  = (r < 3 && c < 3) ? Kinv[r * 4 + c]
                                            : ((r == 3 && c == 3) ? 1.0f : 0.0f);
            }
        for (int r = 0; r < 4; ++r)
            for (int c = 0; c < 4; ++c) {
                float s = 0.0f;
                for (int k = 0; k < 4; ++k) s += T4[r][k] * J[k][c];
                W1[r][c] = s;
            }
        for (int r = 0; r < 3; ++r)
            for (int c = 0; c < 4; ++c) {
                float s = 0.0f;
                for (int k = 0; k < 4; ++k) s += Kb[r * 4 + k] * W1[k][c];
                G[r * 4 + c] = s;
            }
    }
#endif
}

// ---------------------------------------------------------------------------
// Hot kernel: 4 consecutive pixels per thread.
//   depth: one b128 non-temporal load; output: 3x b128 non-temporal stores
//   (streamed data bypasses far cache -> src_img (94MB) stays resident in
//    the 192MB L2, so the 12 bilinear gathers per group are L2 hits).
// ---------------------------------------------------------------------------
__global__ __launch_bounds__(THREADS) void warp_main(
    const float* __restrict__ src,    // [B,C,H,W]
    const float* __restrict__ depth,  // [B,1,H,W]
    const float* __restrict__ G,      // 12 floats (uniform -> s_load)
    float* __restrict__ out)          // [B,C,H,W]
{
    // Uniform projection matrix (scalarized by the compiler).
    const float g00 = G[0], g01 = G[1],  g02 = G[2],  g03 = G[3];
    const float g10 = G[4], g11 = G[5],  g12 = G[6],  g13 = G[7];
    const float g20 = G[8], g21 = G[9],  g22 = G[10], g23 = G[11];

    const unsigned t    = blockIdx.x * THREADS + threadIdx.x;
    const unsigned pix0 = t * PIX_PER_THREAD;
    const unsigned b    = pix0 / HWn;
    const unsigned rem  = pix0 - b * HWn;       // offset within [H,W] plane
    const unsigned y    = rem / Wn;
    const unsigned x0p  = rem - y * Wn;

    const float* __restrict__ srcb = src + (size_t)b * Cn * HWn;

    // Warm L2 for this thread's own src rows (global_prefetch_b8; every pixel
    // is some peer thread's gather target under a near-identity warp).
    __builtin_prefetch(srcb + 0 * HWn + rem, 0, 1);
    __builtin_prefetch(srcb + 1 * HWn + rem, 0, 1);
    __builtin_prefetch(srcb + 2 * HWn + rem, 0, 1);

    // Streamed, read-once depth -> non-temporal b128 load.
    const v4f d4 = __builtin_nontemporal_load((const v4f*)(depth + pix0));

    const float yf = (float)y;
    float r0[PIX_PER_THREAD], r1[PIX_PER_THREAD], r2[PIX_PER_THREAD];

#pragma unroll
    for (int i = 0; i < PIX_PER_THREAD; ++i) {
        const float xf  = (float)(x0p + i);
        const float dep = d4[i];

        // cp = G * [x*d, y*d, d, 1]
        float cx = fmaf(g00, xf, fmaf(g01, yf, g02));
        float cy = fmaf(g10, xf, fmaf(g11, yf, g12));
        float cz = fmaf(g20, xf, fmaf(g21, yf, g22));
        cx = fmaf(cx, dep, g03);
        cy = fmaf(cy, dep, g13);
        cz = fmaf(cz, dep, g23);

        const float inv = 1.0f / (cz + EPS_F);
        const float px = cx * inv;
        const float py = cy * inv;

        // Reference grid mapping (align_corners=False), then unnormalize.
        const float gx = px * (1.0f / (float)(Wn - 1));
        const float gy = py * (1.0f / (float)(Hn - 1));
        const float gridx = (gx - 0.5f) * 2.0f;
        const float gridy = (gy - 0.5f) * 2.0f;
        float ix = ((gridx + 1.0f) * (float)Wn - 1.0f) * 0.5f;
        float iy = ((gridy + 1.0f) * (float)Hn - 1.0f) * 0.5f;
        ix = fminf(fmaxf(ix, 0.0f), (float)(Wn - 1));   // border padding
        iy = fminf(fmaxf(iy, 0.0f), (float)(Hn - 1));

        const float fx0 = floorf(ix);
        const float fy0 = floorf(iy);
        const float wx1 = ix - fx0, wx0 = 1.0f - wx1;
        const float wy1 = iy - fy0, wy0 = 1.0f - wy1;

        const int xi0 = (int)fx0;
        const int yi0 = (int)fy0;
        const int xi1 = min(xi0 + 1, Wn - 1);
        const int yi1 = min(yi0 + 1, Hn - 1);

        const int i00 = yi0 * Wn + xi0;
        const int i01 = yi0 * Wn + xi1;
        const int i10 = yi1 * Wn + xi0;
        const int i11 = yi1 * Wn + xi1;

        const float wa = wy0 * wx0, wb = wy0 * wx1;
        const float wc = wy1 * wx0, wd = wy1 * wx1;

        {
            const float* p = srcb;               // channel 0
            r0[i] = fmaf(wa, p[i00], fmaf(wb, p[i01], fmaf(wc, p[i10], wd * p[i11])));
        }
        {
            const float* p = srcb + HWn;         // channel 1
            r1[i] = fmaf(wa, p[i00], fmaf(wb, p[i01], fmaf(wc, p[i10], wd * p[i11])));
        }
        {
            const float* p = srcb + 2 * HWn;     // channel 2
            r2[i] = fmaf(wa, p[i00], fmaf(wb, p[i01], fmaf(wc, p[i10], wd * p[i11])));
        }
    }

    // Streamed output: non-temporal b128 stores, one per channel.
    float* __restrict__ ob = out + (size_t)b * Cn * HWn + rem;
    v4f o0 = {r0[0], r0[1], r0[2], r0[3]};
    v4f o1 = {r1[0], r1[1], r1[2], r1[3]};
    v4f o2 = {r2[0], r2[1], r2[2], r2[3]};
    __builtin_nontemporal_store(o0, (v4f*)(ob + 0 * HWn));
    __builtin_nontemporal_store(o1, (v4f*)(ob + 1 * HWn));
    __builtin_nontemporal_store(o2, (v4f*)(ob + 2 * HWn));
}

// ---------------------------------------------------------------------------
// Inputs (setup_inputs order): src_img, depth_map, pose, K_batch, K_inv.
// d_ws: 12 floats for G.
// ---------------------------------------------------------------------------
extern "C" void kernel_launch(void* const* d_in, const int* in_sizes, int n_in,
                              void* d_out, int out_size, void* d_ws, size_t ws_size,
                              hipStream_t stream) {
    const float* src  = (const float*)d_in[0];
    const float* dep  = (const float*)d_in[1];
    const float* pose = (const float*)d_in[2];
    const float* Kb   = (const float*)d_in[3];
    const float* Kinv = (const float*)d_in[4];
    float* G   = (float*)d_ws;
    float* dst = (float*)d_out;

    warp_setup<<<1, 32, 0, stream>>>(pose, Kb, Kinv, G);
    warp_main<<<BLOCKS, THREADS, 0, stream>>>(src, dep, G, dst);
}